// MinimalRNNCell_21242908246504
// MI455X (gfx1250) — compile-verified
//
#include <hip/hip_runtime.h>

typedef __attribute__((ext_vector_type(2))) float v2f;
typedef __attribute__((ext_vector_type(8))) float v8f;
typedef __attribute__((ext_vector_type(4))) int   v4i;

#define WAVE 32
#define HP   520   // LDS row stride (floats): 520 % 64 = 8 -> bank-shifted rows
#define KC   32    // K-chunk staged per LDS buffer in phase 1
#define BLW  68    // LDS row stride (floats) for staged B panel (64 cols + 4 pad)

// Async global->LDS path (CDNA5): guarded, with a portable fallback.
#if defined(__has_builtin)
#  if __has_builtin(__builtin_amdgcn_global_load_async_to_lds_b128)
#    define HAVE_ASYNC_LDS 1
#  endif
#endif
#ifndef HAVE_ASYNC_LDS
#  define HAVE_ASYNC_LDS 0
#endif

#if HAVE_ASYNC_LDS
typedef __attribute__((address_space(1))) v4i gv4i;   // global
typedef __attribute__((address_space(3))) v4i lv4i;   // LDS
#endif

__device__ __forceinline__ void wait_async_zero() {
#if defined(__has_builtin)
#  if __has_builtin(__builtin_amdgcn_s_wait_asynccnt)
  __builtin_amdgcn_s_wait_asynccnt(0);
  return;
#  endif
#endif
#if HAVE_ASYNC_LDS
  asm volatile("s_wait_asynccnt 0" ::: "memory");
#endif
}

__device__ __forceinline__ void async_copy_b128(const float* gsrc, float* ldst) {
#if HAVE_ASYNC_LDS
  __builtin_amdgcn_global_load_async_to_lds_b128((gv4i*)const_cast<float*>(gsrc),
                                                 (lv4i*)ldst, 0, 0);
#else
  *(float4*)ldst = *(const float4*)gsrc;
#endif
}

// ---------------------------------------------------------------------------
// Phase 1: XK = X @ K   (X:[M,D], K:[D,U], XK:[M,U], fp32 row-major, M = B*T)
// Block = 256 thr = 8 waves; block tile 128(M) x 64(N); wave strip 16 x 64.
// B panel (KC x 64) double-buffered in LDS via async b128 copies (ASYNCcnt)
// overlapping the WMMA compute on the other buffer.
// WMMA: v_wmma_f32_16x16x4_f32 (fp32 in / fp32 accum).
// ---------------------------------------------------------------------------
__global__ __launch_bounds__(256)
void rnn_xk_gemm(const float* __restrict__ X, const float* __restrict__ Kw,
                 float* __restrict__ XK, int M, int D, int U)
{
  __shared__ float Bl[2 * KC * BLW];   // ~17 KB

  const int tid  = threadIdx.x;
  const int wave = tid / WAVE;          // 0..7
  const int lane = tid % WAVE;
  const int lm   = lane & 15;
  const int lk   = (lane >> 4) * 2;     // 0 or 2

  const int nBlocks = U / 64;
  const int mb = blockIdx.x / nBlocks;
  const int nb = blockIdx.x % nBlocks;

  const int row0 = mb * 128 + wave * 16;
  const int col0 = nb * 64;

  // Cooperative staging: KC*64 floats = 512 b128 units, 2 per thread.
  const int u0r = (tid) >> 4,        u0c = ((tid) & 15) * 4;
  const int u1r = (tid + 256) >> 4,  u1c = ((tid + 256) & 15) * 4;

  auto stage_issue = [&](int kbase, int buf) {
    const float* g0 = Kw + (size_t)(kbase + u0r) * U + col0 + u0c;
    const float* g1 = Kw + (size_t)(kbase + u1r) * U + col0 + u1c;
    float* l0 = &Bl[buf * (KC * BLW) + u0r * BLW + u0c];
    float* l1 = &Bl[buf * (KC * BLW) + u1r * BLW + u1c];
    async_copy_b128(g0, l0);
    async_copy_b128(g1, l1);
  };

  v8f acc0 = {}, acc1 = {}, acc2 = {}, acc3 = {};

  const float* xrow = X + (size_t)(row0 + lm) * D;
  const int nch = D / KC;

  stage_issue(0, 0);
  wait_async_zero();
  __syncthreads();

  for (int c = 0; c < nch; ++c) {
    const int buf = c & 1;
    if (c + 1 < nch)
      stage_issue((c + 1) * KC, (c + 1) & 1);   // async into other buffer

    const float* xck = xrow + c * KC;           // A source (global, cached)
    const float* bl  = &Bl[buf * (KC * BLW)];   // B source (LDS)

    #pragma unroll
    for (int kk = 0; kk < KC; kk += 4) {
      v2f a = *(const v2f*)(xck + kk + lk);
      const float* b0 = bl + (size_t)(kk + lk) * BLW + lm;
      const float* b1 = b0 + BLW;
      v2f b;
      b.x = b0[0];  b.y = b1[0];
      acc0 = __builtin_amdgcn_wmma_f32_16x16x4_f32(false, a, false, b, (short)0, acc0, false, false);
      b.x = b0[16]; b.y = b1[16];
      acc1 = __builtin_amdgcn_wmma_f32_16x16x4_f32(false, a, false, b, (short)0, acc1, false, false);
      b.x = b0[32]; b.y = b1[32];
      acc2 = __builtin_amdgcn_wmma_f32_16x16x4_f32(false, a, false, b, (short)0, acc2, false, false);
      b.x = b0[48]; b.y = b1[48];
      acc3 = __builtin_amdgcn_wmma_f32_16x16x4_f32(false, a, false, b, (short)0, acc3, false, false);
    }

    wait_async_zero();
    __syncthreads();
  }

  const int rbase = row0 + ((lane >> 4) << 3);
  float* outp = XK + (size_t)rbase * U + col0 + lm;
  #pragma unroll
  for (int r = 0; r < 8; ++r) {
    float* o = outp + (size_t)r * U;
    o[0]  = acc0[r];
    o[16] = acc1[r];
    o[32] = acc2[r];
    o[48] = acc3[r];
  }
}

// ---------------------------------------------------------------------------
// Phase 2: sequential scan  h_t = xk_t + h_{t-1} @ R
// KEY: batch rows are independent across the scan -> split B=64 into B/16
// independent workgroups of 16 rows each (no inter-WG sync, 4x parallelism).
// Each WG: 512 thr = 16 waves on its own WGP; its 16-row hidden state is
// double-buffered in LDS (2 x 16 x HP floats ~ 66 KB). R stays L2-resident.
// Per wave: one 16x32 strip (2 WMMA tiles); software-pipelined inner loop
// (A frag via ds_load_b64 of h_prev, R frags via b32 loads, one step ahead).
// ---------------------------------------------------------------------------
__global__ __launch_bounds__(512)
void rnn_scan(const float* __restrict__ R,    // [U, U]
              const float* __restrict__ h0,   // [B, U]
              float* __restrict__ out,        // [B, T, U], pre-filled with XK
              int T, int U)
{
  extern __shared__ float hbuf[];             // [2][16][HP]
  const int tid    = threadIdx.x;
  const int wave   = tid / WAVE;              // 0..15
  const int lane   = tid % WAVE;
  const int lm     = lane & 15;
  const int lkHalf = lane >> 4;               // 0/1
  const int lk     = lkHalf * 2;

  const int rowg = blockIdx.x * 16;           // global batch-row base of this WG

  // h0 rows [rowg, rowg+16) -> buffer 0.
  for (int i = tid; i < (16 * U) / 4; i += (int)blockDim.x) {
    const int r = (i * 4) / U;
    const int c = (i * 4) % U;
    *(float4*)&hbuf[(size_t)r * HP + c] =
        *(const float4*)&h0[(size_t)(rowg + r) * U + c];
  }
  __syncthreads();

  const int col0 = wave * 32;                 // 2 tiles: cols [col0, col0+32)

  for (int t = 0; t < T; ++t) {
    const float* hprev = hbuf + (size_t)(t & 1) * 16 * HP;
    float*       hcur  = hbuf + (size_t)((t + 1) & 1) * 16 * HP;

    v8f acc0 = {}, acc1 = {};

    const float* hrow = hprev + (size_t)lm * HP;
    const float* Rcol = R + col0 + lm;

    // Prime the pipeline with fragments for k = 0.
    v2f a_cur = *(const v2f*)(hrow + lk);
    v2f bc0, bc1;
    {
      const float* r0 = Rcol + (size_t)lk * U;
      const float* r1 = r0 + U;
      bc0.x = r0[0];  bc0.y = r1[0];
      bc1.x = r0[16]; bc1.y = r1[16];
    }

    for (int k = 0; k < U; k += 4) {
      if ((k & 127) == 0 && k + 128 < U)
        __builtin_prefetch(Rcol + (size_t)(k + 128) * U, 0, 0);

      // Issue next step's loads (branchless; last iteration reloads itself).
      const int kn = (k + 4 < U) ? (k + 4) : k;
      v2f a_nxt = *(const v2f*)(hrow + kn + lk);           // ds_load_b64
      const float* r0 = Rcol + (size_t)(kn + lk) * U;
      const float* r1 = r0 + U;
      v2f bn0, bn1;
      bn0.x = r0[0];  bn0.y = r1[0];
      bn1.x = r0[16]; bn1.y = r1[16];

      acc0 = __builtin_amdgcn_wmma_f32_16x16x4_f32(false, a_cur, false, bc0, (short)0, acc0, false, false);
      acc1 = __builtin_amdgcn_wmma_f32_16x16x4_f32(false, a_cur, false, bc1, (short)0, acc1, false, false);

      a_cur = a_nxt;
      bc0 = bn0; bc1 = bn1;
    }

    // h_t = acc + xk_t (xk already in `out`); write out + LDS h buffer.
    const int rbase = lkHalf << 3;
    #pragma unroll
    for (int r = 0; r < 8; ++r) {
      const int rl = rbase + r;                           // local row 0..15
      float* op = out + ((size_t)(rowg + rl) * T + t) * U + col0 + lm;
      float* hp = hcur + (size_t)rl * HP + col0 + lm;
      float v0 = acc0[r] + op[0];
      float v1 = acc1[r] + op[16];
      op[0]  = v0;  hp[0]  = v0;
      op[16] = v1;  hp[16] = v1;
    }

    __syncthreads();   // this WG's h_t fully in LDS before step t+1
  }
}

// ---------------------------------------------------------------------------
extern "C" void kernel_launch(void* const* d_in, const int* in_sizes, int n_in,
                              void* d_out, int out_size, void* d_ws, size_t ws_size,
                              hipStream_t stream) {
  const float* x  = (const float*)d_in[0];   // [B, T, D]
  const float* h0 = (const float*)d_in[1];   // [B, U]
  const float* Kw = (const float*)d_in[2];   // [D, U]
  const float* Rw = (const float*)d_in[3];   // [U, U]
  float* out = (float*)d_out;                // [B, T, U]

  // Recover dims from flat sizes: U*U, D*U, B*U, B*T*D.
  int U = 1;
  while ((long long)(U + 1) * (U + 1) <= (long long)in_sizes[3]) ++U;
  const int D = in_sizes[2] / U;
  const int B = in_sizes[1] / U;
  const int M = in_sizes[0] / D;             // B*T
  const int T = M / B;

  // Phase 1: XK = X @ K written straight into d_out.
  const int grid1 = (M / 128) * (U / 64);
  rnn_xk_gemm<<<grid1, 256, 0, stream>>>(x, Kw, out, M, D, U);

  // Phase 2: batch-parallel scan, one 16-row WG per WGP.
  const int grid2 = B / 16;                  // independent batch slices
  const size_t ldsBytes = (size_t)2 * 16 * HP * sizeof(float);  // ~66 KB
  rnn_scan<<<grid2, 512, ldsBytes, stream>>>(Rw, h0, out, T, U);
}